// MultiModalEncoder_5738076308108
// MI455X (gfx1250) — compile-verified
//
#include <hip/hip_runtime.h>
#include <math.h>

typedef __attribute__((ext_vector_type(2))) float v2f;
typedef __attribute__((ext_vector_type(8))) float v8f;

#if defined(__has_builtin)
#if __has_builtin(__builtin_amdgcn_global_load_async_to_lds_b32) && \
    __has_builtin(__builtin_amdgcn_s_wait_asynccnt)
#define HAVE_ASYNC_LDS 1
#endif
#endif

typedef __attribute__((address_space(1))) int* gint_p;
typedef __attribute__((address_space(3))) int* lint_p;

// D = A(16x4) * B(4x16) + C, fp32 WMMA (exact-precision CDNA5 matrix path)
__device__ __forceinline__ v8f wmma4(v2f a, v2f b, v8f c) {
  return __builtin_amdgcn_wmma_f32_16x16x4_f32(false, a, false, b, (short)0, c,
                                               false, false);
}

// ---------------------------------------------------------------------------
// Generic fp32 WMMA GEMM: C[b,m,n] = sum_k A[b,m,k]*B[b,k,n] (+biasRow[m])
//                         (+biasCol[n]) (+res[b*rB + m*rR + n*rC])
// One wave per 16x16 tile. A row-major (lda), B row-major (ldb).
// B address is pointer-stepped (bptr += 4*ldb) to keep the k-loop free of
// 64-bit multiplies.
// ---------------------------------------------------------------------------
__global__ void k_gemm16(const float* __restrict__ A, int lda, long long bA,
                         const float* __restrict__ Bm, int ldb, long long bB,
                         float* __restrict__ Cm, int ldc, long long bC,
                         const float* __restrict__ biasRow,
                         const float* __restrict__ biasCol,
                         const float* __restrict__ res, long long rB, int rR,
                         int rC, int K) {
  const int lane = threadIdx.x & 31;
  const int n16 = lane & 15, hi = lane >> 4;
  const int m0 = blockIdx.y * 16, n0 = blockIdx.x * 16;
  const int b = blockIdx.z;
  const float* arow = A + (long long)b * bA + (long long)(m0 + n16) * lda + 2 * hi;
  const float* bptr = Bm + (long long)b * bB + (long long)(2 * hi) * ldb + n0 + n16;
  const long long bstep = (long long)4 * ldb;
  v8f acc = {};
  for (int k0 = 0; k0 < K; k0 += 4) {
    if ((k0 & 31) == 0) {
      __builtin_prefetch(arow + k0 + 32, 0, 3);   // global_prefetch_b8
      __builtin_prefetch(bptr + 2 * bstep, 0, 3);
    }
    v2f a;
    a.x = arow[k0];
    a.y = arow[k0 + 1];
    v2f bb;
    bb.x = bptr[0];
    bb.y = bptr[ldb];
    acc = wmma4(a, bb, acc);
    bptr += bstep;
  }
  const int n = n0 + n16;
  const float bc = biasCol ? biasCol[n] : 0.f;
#pragma unroll
  for (int r = 0; r < 8; ++r) {
    const int m = m0 + r + 8 * hi;
    float v = acc[r] + bc;
    if (biasRow) v += biasRow[m];
    if (res) v += res[(long long)b * rB + (long long)m * rR + (long long)n * rC];
    Cm[(long long)b * bC + (long long)m * ldc + n] = v;
  }
}

// ---------------------------------------------------------------------------
// Modality embedding: e = silu(table[idx] @ w1 + b1) @ w2 + b2   (idx = blockIdx.x)
// ---------------------------------------------------------------------------
__global__ void k_modality(const float* __restrict__ table,
                           const float* __restrict__ w1, const float* __restrict__ b1,
                           const float* __restrict__ w2, const float* __restrict__ b2,
                           float* __restrict__ e) {
  const int idx = blockIdx.x, j = threadIdx.x;
  __shared__ float se[128], sh[128];
  se[j] = table[idx * 128 + j];
  __syncthreads();
  float a = b1[j];
  for (int i = 0; i < 128; ++i) a += se[i] * w1[i * 128 + j];
  sh[j] = a / (1.f + __expf(-a));  // silu
  __syncthreads();
  float o = b2[j];
  for (int i = 0; i < 128; ++i) o += sh[i] * w2[i * 128 + j];
  e[idx * 128 + j] = o;
}

__global__ void k_biasmix(const float* __restrict__ e, const float* __restrict__ ib,
                          const float* __restrict__ pbv, float* __restrict__ cb,
                          float* __restrict__ pb) {
  const int i = threadIdx.x;
  cb[i] = ib[i] + e[i];          // img_in_b + e_img
  pb[i] = pbv[i] + e[128 + i];   // prop_in_b + e_prop
}

// ---------------------------------------------------------------------------
// 3x3 conv, 3->128, pad 1, stride 1 on 48x48. One block per (b,co).
// Whole input image (3x48x48 = 27.6KB) staged into LDS — via the CDNA5
// async global->LDS path (ASYNCcnt) when the builtin is available.
// bias already includes the modality embedding.
// ---------------------------------------------------------------------------
__global__ void k_conv_in(const float* __restrict__ img, const float* __restrict__ w,
                          const float* __restrict__ bias, float* __restrict__ out) {
  const int b = blockIdx.x >> 7, co = blockIdx.x & 127;
  __shared__ float sIn[3 * 48 * 48];
  __shared__ float sw[27];
#if HAVE_ASYNC_LDS
  {
    const float* gsrc = img + (long long)b * 6912;
    for (int i = threadIdx.x; i < 6912; i += 256) {
      __builtin_amdgcn_global_load_async_to_lds_b32(
          (gint_p)(gsrc + i), (lint_p)(sIn + i), 0, 0);
    }
    __builtin_amdgcn_s_wait_asynccnt(0);
  }
#else
  for (int i = threadIdx.x; i < 6912; i += 256) sIn[i] = img[(long long)b * 6912 + i];
#endif
  if (threadIdx.x < 27) sw[threadIdx.x] = w[co * 27 + threadIdx.x];
  __syncthreads();
  const float bb = bias[co];
  for (int p = threadIdx.x; p < 2304; p += 256) {
    const int y = p / 48, x = p - y * 48;
    float acc = bb;
#pragma unroll
    for (int ci = 0; ci < 3; ++ci)
#pragma unroll
      for (int dy = 0; dy < 3; ++dy) {
        const int iy = y + dy - 1;
        if (iy < 0 || iy >= 48) continue;
#pragma unroll
        for (int dx = 0; dx < 3; ++dx) {
          const int ix = x + dx - 1;
          if (ix < 0 || ix >= 48) continue;
          acc += sIn[ci * 2304 + iy * 48 + ix] * sw[ci * 9 + dy * 3 + dx];
        }
      }
    out[((long long)b * 128 + co) * 2304 + p] = acc;
  }
}

// ---------------------------------------------------------------------------
// GroupNorm: 32 groups over (B,128,L). One block per (b,g), group = 4*L elems.
// ---------------------------------------------------------------------------
__global__ void k_groupnorm(const float* __restrict__ x, const float* __restrict__ g,
                            const float* __restrict__ be, float* __restrict__ y,
                            int L) {
  const int b = blockIdx.x >> 5, grp = blockIdx.x & 31;
  const float* xb = x + ((long long)b * 128 + grp * 4) * L;
  float* yb = y + ((long long)b * 128 + grp * 4) * L;
  const int n = 4 * L;
  __shared__ float s1[256], s2[256];
  float a = 0.f, q = 0.f;
  for (int i = threadIdx.x; i < n; i += 256) {
    const float v = xb[i];
    a += v;
    q += v * v;
  }
  s1[threadIdx.x] = a;
  s2[threadIdx.x] = q;
  __syncthreads();
  for (int off = 128; off; off >>= 1) {
    if (threadIdx.x < off) {
      s1[threadIdx.x] += s1[threadIdx.x + off];
      s2[threadIdx.x] += s2[threadIdx.x + off];
    }
    __syncthreads();
  }
  const float mu = s1[0] / n;
  const float rs = rsqrtf(s2[0] / n - mu * mu + 1e-5f);
  for (int i = threadIdx.x; i < n; i += 256) {
    const int cc = grp * 4 + i / L;
    yb[i] = (xb[i] - mu) * rs * g[cc] + be[cc];
  }
}

// ---------------------------------------------------------------------------
// LayerNorm over C=128 per token; strided gather input, token-major output.
// ---------------------------------------------------------------------------
__global__ void k_layernorm(const float* __restrict__ x, long long sB, long long sT,
                            int sC, const float* __restrict__ g,
                            const float* __restrict__ be, float* __restrict__ y,
                            int Ntok) {
  const int tok = blockIdx.x;
  const int b = tok / Ntok, t = tok - b * Ntok;
  const int c = threadIdx.x;
  __shared__ float s1[128], s2[128];
  const float v = x[(long long)b * sB + (long long)t * sT + (long long)c * sC];
  s1[c] = v;
  s2[c] = v * v;
  __syncthreads();
  for (int off = 64; off; off >>= 1) {
    if (c < off) {
      s1[c] += s1[c + off];
      s2[c] += s2[c + off];
    }
    __syncthreads();
  }
  const float mu = s1[0] * (1.f / 128.f);
  const float rs = rsqrtf(s2[0] * (1.f / 128.f) - mu * mu + 1e-5f);
  y[(long long)tok * 128 + c] = (v - mu) * rs * g[c] + be[c];
}

// ---------------------------------------------------------------------------
// Flash attention, spatial layout: qkv is (B, 3*128, L), q/k/v row = part*128+h*32+dd.
// Per wave: 16 queries for one (b,h). S^T = K^T*Q via WMMA so the softmax axis
// lands in the C-tile component dim (row stats = 7 VALU max + 1 shfl_xor(16)).
// P is re-fragmented to WMMA-B layout with wave-local shuffles; O^T = V*P^T.
// ---------------------------------------------------------------------------
__device__ __forceinline__ void flash_tile(v8f s, float& mrun, float& lrun, v8f& o0,
                                           v8f& o1, const float* vbase,
                                           long long vRow, int sA0, int n16,
                                           int hi) {
  float mloc = s[0];
#pragma unroll
  for (int r = 1; r < 8; ++r) mloc = fmaxf(mloc, s[r]);
  mloc = fmaxf(mloc, __shfl_xor(mloc, 16, 32));
  const float mnew = fmaxf(mrun, mloc);
  v8f p;
  float lt = 0.f;
#pragma unroll
  for (int r = 0; r < 8; ++r) {
    p[r] = __expf(s[r] - mnew);
    lt += p[r];
  }
  lt += __shfl_xor(lt, 16, 32);
  const float corr = __expf(mrun - mnew);
  lrun = lrun * corr + lt;
  mrun = mnew;
#pragma unroll
  for (int r = 0; r < 8; ++r) {
    o0[r] *= corr;
    o1[r] *= corr;
  }
  // K-chunks of 4 over the 16 keys of this tile
#pragma unroll
  for (int kc = 0; kc < 4; ++kc) {
    const int src = n16 + ((kc >= 2) ? 16 : 0);
    const float x00 = __shfl(p[(kc * 4 + 0) & 7], src, 32);
    const float x01 = __shfl(p[(kc * 4 + 2) & 7], src, 32);
    const float x10 = __shfl(p[(kc * 4 + 1) & 7], src, 32);
    const float x11 = __shfl(p[(kc * 4 + 3) & 7], src, 32);
    v2f pf;
    pf.x = hi ? x01 : x00;  // B[k=kc*4+2*hi+0][n]
    pf.y = hi ? x11 : x10;  // B[k=kc*4+2*hi+1][n]
    const long long sA = sA0 + kc * 4 + 2 * hi;
    v2f a0, a1;
    a0.x = vbase[(long long)n16 * vRow + sA];           // dd tile 0 (strided layout)
    a0.y = vbase[(long long)n16 * vRow + sA + 1];
    a1.x = vbase[(long long)(16 + n16) * vRow + sA];    // dd tile 1
    a1.y = vbase[(long long)(16 + n16) * vRow + sA + 1];
    o0 = wmma4(a0, pf, o0);
    o1 = wmma4(a1, pf, o1);
  }
}

__global__ void k_attn_spatial(const float* __restrict__ qkv, float* __restrict__ o,
                               int L, float scale) {
  const int lane = threadIdx.x & 31;
  const int n16 = lane & 15, hi = lane >> 4;
  const int t0 = blockIdx.x * 16;
  const int b = blockIdx.y >> 2, h = blockIdx.y & 3;
  const float* qb = qkv + ((long long)b * 384 + h * 32) * L;
  const float* kb = qb + (long long)128 * L;
  const float* vb = qb + (long long)256 * L;
  v2f qf[8];
#pragma unroll
  for (int kc = 0; kc < 8; ++kc) {  // B fragments of Q, pre-scaled
    const int dd = kc * 4 + 2 * hi;
    qf[kc].x = qb[(long long)dd * L + t0 + n16] * scale;
    qf[kc].y = qb[(long long)(dd + 1) * L + t0 + n16] * scale;
  }
  v8f o0 = {}, o1 = {};
  float mrun = -1e30f, lrun = 0.f;
  for (int st = 0; st < L / 16; ++st) {
    const int s0 = st * 16;
    v8f s = {};
#pragma unroll
    for (int kc = 0; kc < 8; ++kc) {  // S^T tile: A = K^T fragment
      const int dd = kc * 4 + 2 * hi;
      v2f a;
      a.x = kb[(long long)dd * L + s0 + n16];
      a.y = kb[(long long)(dd + 1) * L + s0 + n16];
      s = wmma4(a, qf[kc], s);
    }
    flash_tile(s, mrun, lrun, o0, o1, vb, L, s0, n16, hi);
  }
  const float inv = 1.f / lrun;
  float* ob = o + ((long long)b * 128 + h * 32) * L + t0 + n16;
#pragma unroll
  for (int r = 0; r < 8; ++r) {
    const int dd = r + 8 * hi;
    ob[(long long)dd * L] = o0[r] * inv;
    ob[(long long)(16 + dd) * L] = o1[r] * inv;
  }
}

// ---------------------------------------------------------------------------
// Flash cross-attention: queries = image tokens (B,576,384 token-major qkv),
// keys/vals = concat(x-kv[576], ctx-kv[512]) where ckv is (B,512,384) and the
// reference uses ckv cols [0,128) as K and [128,256) as V. Output token-major.
// ---------------------------------------------------------------------------
__device__ __forceinline__ void flash_tile_tok(v8f s, float& mrun, float& lrun,
                                               v8f& o0, v8f& o1,
                                               const float* vbase, int srow,
                                               int n16, int hi) {
  float mloc = s[0];
#pragma unroll
  for (int r = 1; r < 8; ++r) mloc = fmaxf(mloc, s[r]);
  mloc = fmaxf(mloc, __shfl_xor(mloc, 16, 32));
  const float mnew = fmaxf(mrun, mloc);
  v8f p;
  float lt = 0.f;
#pragma unroll
  for (int r = 0; r < 8; ++r) {
    p[r] = __expf(s[r] - mnew);
    lt += p[r];
  }
  lt += __shfl_xor(lt, 16, 32);
  const float corr = __expf(mrun - mnew);
  lrun = lrun * corr + lt;
  mrun = mnew;
#pragma unroll
  for (int r = 0; r < 8; ++r) {
    o0[r] *= corr;
    o1[r] *= corr;
  }
#pragma unroll
  for (int kc = 0; kc < 4; ++kc) {
    const int src = n16 + ((kc >= 2) ? 16 : 0);
    const float x00 = __shfl(p[(kc * 4 + 0) & 7], src, 32);
    const float x01 = __shfl(p[(kc * 4 + 2) & 7], src, 32);
    const float x10 = __shfl(p[(kc * 4 + 1) & 7], src, 32);
    const float x11 = __shfl(p[(kc * 4 + 3) & 7], src, 32);
    v2f pf;
    pf.x = hi ? x01 : x00;
    pf.y = hi ? x11 : x10;
    const long long sA = srow + kc * 4 + 2 * hi;
    v2f a0, a1;  // A[m=dd][k=s] = V[s][dd], token-major (row stride 384)
    a0.x = vbase[sA * 384 + n16];
    a0.y = vbase[(sA + 1) * 384 + n16];
    a1.x = vbase[sA * 384 + 16 + n16];
    a1.y = vbase[(sA + 1) * 384 + 16 + n16];
    o0 = wmma4(a0, pf, o0);
    o1 = wmma4(a1, pf, o1);
  }
}

__global__ void k_attn_cross(const float* __restrict__ qkvx,
                             const float* __restrict__ ckv, float* __restrict__ o,
                             float scale) {
  const int T = 576, S2 = 512;
  const int lane = threadIdx.x & 31;
  const int n16 = lane & 15, hi = lane >> 4;
  const int t0 = blockIdx.x * 16;
  const int b = blockIdx.y >> 2, h = blockIdx.y & 3;
  const float* qb = qkvx + (long long)b * T * 384 + h * 32;
  v2f qf[8];
#pragma unroll
  for (int kc = 0; kc < 8; ++kc) {
    const int dd = kc * 4 + 2 * hi;
    const float* qr = qb + (long long)(t0 + n16) * 384 + dd;
    qf[kc].x = qr[0] * scale;
    qf[kc].y = qr[1] * scale;
  }
  v8f o0 = {}, o1 = {};
  float mrun = -1e30f, lrun = 0.f;
  for (int st = 0; st < (T + S2) / 16; ++st) {
    const int s0 = st * 16;
    const bool fromX = (s0 < T);
    const float* kbase = fromX ? qkvx + (long long)b * T * 384 + 128 + h * 32
                               : ckv + (long long)b * S2 * 384 + h * 32;
    const float* vbase = fromX ? qkvx + (long long)b * T * 384 + 256 + h * 32
                               : ckv + (long long)b * S2 * 384 + 128 + h * 32;
    const int srow = fromX ? s0 : s0 - T;
    v8f s = {};
#pragma unroll
    for (int kc = 0; kc < 8; ++kc) {
      const int dd = kc * 4 + 2 * hi;
      const float* kr = kbase + (long long)(srow + n16) * 384 + dd;
      v2f a;
      a.x = kr[0];
      a.y = kr[1];
      s = wmma4(a, qf[kc], s);
    }
    flash_tile_tok(s, mrun, lrun, o0, o1, vbase, srow, n16, hi);
  }
  const float inv = 1.f / lrun;
  float* ob = o + ((long long)b * T + t0 + n16) * 128 + h * 32;
#pragma unroll
  for (int r = 0; r < 8; ++r) {
    const int dd = r + 8 * hi;
    ob[dd] = o0[r] * inv;
    ob[16 + dd] = o1[r] * inv;
  }
}

// ---------------------------------------------------------------------------
// Stride-2 3x3 conv 128->128 as implicit GEMM: M=co(128), N=24*24(576),
// K=128*9(1152). A = w row-major (contiguous), B gathered on the fly with pad.
// ---------------------------------------------------------------------------
__global__ void k_conv_ds(const float* __restrict__ x, const float* __restrict__ w,
                          const float* __restrict__ bias, float* __restrict__ out) {
  const int lane = threadIdx.x & 31;
  const int n16 = lane & 15, hi = lane >> 4;
  const int nt = blockIdx.x, mt = blockIdx.y, b = blockIdx.z;
  const int n = nt * 16 + n16;
  const int oy = n / 24, ox = n - oy * 24;
  const float* xb = x + (long long)b * 128 * 2304;
  const float* arow = w + (long long)(mt * 16 + n16) * 1152 + 2 * hi;
  v8f acc = {};
  for (int k0 = 0; k0 < 1152; k0 += 4) {
    v2f a;
    a.x = arow[k0];
    a.y = arow[k0 + 1];
    v2f bf;
#pragma unroll
    for (int c = 0; c < 2; ++c) {
      const int k = k0 + 2 * hi + c;
      const int ci = k / 9;
      const int r9 = k - ci * 9;
      const int dy = r9 / 3, dx = r9 - dy * 3;
      const int iy = 2 * oy - 1 + dy, ix = 2 * ox - 1 + dx;
      float v = 0.f;
      if (iy >= 0 && iy < 48 && ix >= 0 && ix < 48)
        v = xb[((long long)ci * 48 + iy) * 48 + ix];
      if (c == 0)
        bf.x = v;
      else
        bf.y = v;
    }
    acc = wmma4(a, bf, acc);
  }
#pragma unroll
  for (int r = 0; r < 8; ++r) {
    const int m = mt * 16 + r + 8 * hi;
    out[((long long)b * 128 + m) * 576 + n] = acc[r] + bias[m];
  }
}

// ---------------------------------------------------------------------------
static inline void gemm(hipStream_t st, const float* A, int lda, long long bA,
                        const float* B, int ldb, long long bB, float* C, int ldc,
                        long long bC, const float* bRow, const float* bCol,
                        const float* res, long long rB, int rR, int rC, int M,
                        int N, int K, int batch) {
  dim3 g(N / 16, M / 16, batch);
  k_gemm16<<<g, 32, 0, st>>>(A, lda, bA, B, ldb, bB, C, ldc, bC, bRow, bCol, res,
                             rB, rR, rC, K);
}

extern "C" void kernel_launch(void* const* d_in, const int* in_sizes, int n_in,
                              void* d_out, int out_size, void* d_ws, size_t ws_size,
                              hipStream_t stream) {
  const float* image     = (const float*)d_in[0];
  const float* prop      = (const float*)d_in[1];
  const float* img_in_w  = (const float*)d_in[2];
  const float* img_in_b  = (const float*)d_in[3];
  const float* prop_in_w = (const float*)d_in[4];
  const float* prop_in_b = (const float*)d_in[5];
  const float* modal_tab = (const float*)d_in[6];
  const float* me_w1     = (const float*)d_in[7];
  const float* me_b1     = (const float*)d_in[8];
  const float* me_w2     = (const float*)d_in[9];
  const float* me_b2     = (const float*)d_in[10];
  const float* a2_gn_s   = (const float*)d_in[11];
  const float* a2_gn_b   = (const float*)d_in[12];
  const float* a2_qkv_w  = (const float*)d_in[13];
  const float* a2_qkv_b  = (const float*)d_in[14];
  const float* a2_out_w  = (const float*)d_in[15];
  const float* a2_out_b  = (const float*)d_in[16];
  const float* a1_gn_s   = (const float*)d_in[17];
  const float* a1_gn_b   = (const float*)d_in[18];
  const float* a1_qkv_w  = (const float*)d_in[19];
  const float* a1_qkv_b  = (const float*)d_in[20];
  const float* a1_out_w  = (const float*)d_in[21];
  const float* a1_out_b  = (const float*)d_in[22];
  const float* ds_w      = (const float*)d_in[23];
  const float* ds_b      = (const float*)d_in[24];
  const float* ca_ln_s   = (const float*)d_in[25];
  const float* ca_ln_b   = (const float*)d_in[26];
  const float* ca_qkv_w  = (const float*)d_in[27];
  const float* ca_qkv_b  = (const float*)d_in[28];
  const float* ca_out_w  = (const float*)d_in[29];
  const float* ca_out_b  = (const float*)d_in[30];
  const float* fin_w     = (const float*)d_in[31];
  const float* fin_b     = (const float*)d_in[32];

  const int B = 2, LI = 2304, LP = 512, T = 576;
  const float scale = 0.17677669529663687f;  // 32^-0.5

  float* W = (float*)d_ws;
  size_t off = 0;
  auto alloc = [&](size_t nf) { float* p = W + off; off += nf; return p; };
  float* e    = alloc(256);            // e_img | e_prop
  float* cb   = alloc(128);            // img_in_b + e_img
  float* pb   = alloc(128);            // prop_in_b + e_prop
  float* h1   = alloc((size_t)B * 128 * LI);
  float* xf   = alloc((size_t)B * 128 * LI);
  float* qkvi = alloc((size_t)B * 384 * LI);
  float* oi   = alloc((size_t)B * 128 * LI);
  float* h2   = alloc((size_t)B * 128 * LI);
  float* hds  = alloc((size_t)B * 128 * T);   // (B,128,24,24) == img tokens^T
  float* p0   = alloc((size_t)B * 128 * LP);
  float* xfp  = alloc((size_t)B * 128 * LP);
  float* qkvp = alloc((size_t)B * 384 * LP);
  float* op   = alloc((size_t)B * 128 * LP);
  float* p2   = alloc((size_t)B * 128 * LP);  // ctx^T
  float* lnx  = alloc((size_t)B * T * 128);
  float* lnc  = alloc((size_t)B * LP * 128);
  float* qkvx = alloc((size_t)B * T * 384);
  float* ckv  = alloc((size_t)B * LP * 384);
  float* ox   = alloc((size_t)B * T * 128);
  float* fused= alloc((size_t)B * T * 128);
  (void)ws_size; (void)in_sizes; (void)n_in; (void)out_size;

  // modality embeddings (idx 0 = image, idx 1 = prop) and fused biases
  k_modality<<<2, 128, 0, stream>>>(modal_tab, me_w1, me_b1, me_w2, me_b2, e);
  k_biasmix<<<1, 128, 0, stream>>>(e, img_in_b, prop_in_b, cb, pb);

  // ---- image branch ----
  k_conv_in<<<B * 128, 256, 0, stream>>>(image, img_in_w, cb, h1);
  k_groupnorm<<<B * 32, 256, 0, stream>>>(h1, a2_gn_s, a2_gn_b, xf, LI);
  gemm(stream, a2_qkv_w, 128, 0, xf, LI, (long long)128 * LI, qkvi, LI,
       (long long)384 * LI, a2_qkv_b, nullptr, nullptr, 0, 0, 0, 384, LI, 128, B);
  k_attn_spatial<<<dim3(LI / 16, B * 4), 32, 0, stream>>>(qkvi, oi, LI, scale);
  gemm(stream, a2_out_w, 128, 0, oi, LI, (long long)128 * LI, h2, LI,
       (long long)128 * LI, a2_out_b, nullptr, h1, (long long)128 * LI, LI, 1, 128,
       LI, 128, B);
  k_conv_ds<<<dim3(36, 8, B), 32, 0, stream>>>(h2, ds_w, ds_b, hds);

  // ---- prop branch ----
  gemm(stream, prop_in_w, 16, 0, prop, LP, (long long)16 * LP, p0, LP,
       (long long)128 * LP, pb, nullptr, nullptr, 0, 0, 0, 128, LP, 16, B);
  k_groupnorm<<<B * 32, 256, 0, stream>>>(p0, a1_gn_s, a1_gn_b, xfp, LP);
  gemm(stream, a1_qkv_w, 128, 0, xfp, LP, (long long)128 * LP, qkvp, LP,
       (long long)384 * LP, a1_qkv_b, nullptr, nullptr, 0, 0, 0, 384, LP, 128, B);
  k_attn_spatial<<<dim3(LP / 16, B * 4), 32, 0, stream>>>(qkvp, op, LP, scale);
  gemm(stream, a1_out_w, 128, 0, op, LP, (long long)128 * LP, p2, LP,
       (long long)128 * LP, a1_out_b, nullptr, p0, (long long)128 * LP, LP, 1, 128,
       LP, 128, B);

  // ---- cross-modal attention ----
  k_layernorm<<<B * T, 128, 0, stream>>>(hds, (long long)128 * T, 1, T, ca_ln_s,
                                         ca_ln_b, lnx, T);
  k_layernorm<<<B * LP, 128, 0, stream>>>(p2, (long long)128 * LP, 1, LP, ca_ln_s,
                                          ca_ln_b, lnc, LP);
  gemm(stream, lnx, 128, (long long)T * 128, ca_qkv_w, 384, 0, qkvx, 384,
       (long long)T * 384, nullptr, ca_qkv_b, nullptr, 0, 0, 0, T, 384, 128, B);
  gemm(stream, lnc, 128, (long long)LP * 128, ca_qkv_w, 384, 0, ckv, 384,
       (long long)LP * 384, nullptr, ca_qkv_b, nullptr, 0, 0, 0, LP, 384, 128, B);
  k_attn_cross<<<dim3(T / 16, B * 4), 32, 0, stream>>>(qkvx, ckv, ox, scale);
  // out-proj + residual (residual = img tokens, gathered strided from hds)
  gemm(stream, ox, 128, (long long)T * 128, ca_out_w, 128, 0, fused, 128,
       (long long)T * 128, nullptr, ca_out_b, hds, (long long)128 * T, 1, T, T, 128,
       128, B);

  // ---- final projection ----
  gemm(stream, fused, 128, (long long)T * 128, fin_w, 256, 0, (float*)d_out, 256,
       (long long)T * 256, nullptr, fin_b, nullptr, 0, 0, 0, T, 256, 128, B);
}